// CentroidLossExcludingSelf_58729382806234
// MI455X (gfx1250) — compile-verified
//
#include <hip/hip_runtime.h>

#define B_N 32768
#define D_N 1024
#define C_N 256
#define KSPLIT 8
#define W_LOSS 0.0005f
#define EPS_N 1e-12f

typedef __attribute__((ext_vector_type(2))) float v2f;
typedef __attribute__((ext_vector_type(8))) float v8f;

// ---- workspace layout (byte offsets) ----
#define OFF_INVN   0u          // 32768 f32   (131072 B)
#define OFF_SUMS   131072u     // 256*1024 f32 (1048576 B)
#define OFF_HIST   1179648u    // 256*256 i32 -> becomes exclusive offsets (262144 B)
#define OFF_LRANK  1441792u    // 32768 i32 (131072 B)
#define OFF_EXCL   1572864u    // 32768 i32 (131072 B)
#define OFF_CNT    1703936u    // 256 f32 (1024 B)
#define OFF_PART   1704960u    // 4096 f32 (16384 B)
#define OFF_PSUM   1721344u    // 8*256*1024 f32 (8388608 B) -> total ~9.6 MB

__device__ inline float wave_reduce_add(float v) {
  #pragma unroll
  for (int m = 16; m >= 1; m >>= 1) v += __shfl_xor(v, m, 32);
  return v;
}

// K1: inverse row norms, one wave (32 lanes) per row of 1024
__global__ void k_norms(const float* __restrict__ feat, float* __restrict__ invn) {
  int wave = (blockIdx.x * blockDim.x + threadIdx.x) >> 5;
  int lane = threadIdx.x & 31;
  const float* row = feat + (size_t)wave * D_N;
  float acc = 0.f;
  #pragma unroll
  for (int j = 0; j < 8; ++j) {
    int d = j * 128 + lane * 4;
    float4 v = *(const float4*)(row + d);
    acc += v.x * v.x + v.y * v.y + v.z * v.z + v.w * v.w;
  }
  acc = wave_reduce_add(acc);
  if (lane == 0) invn[wave] = 1.0f / fmaxf(sqrtf(acc), EPS_N);
}

// K2: per-chunk(128) class histogram, in-chunk rank, init excl=-1
__global__ void k_hist(const int* __restrict__ labels, int* __restrict__ hist,
                       int* __restrict__ lrank, int* __restrict__ excl) {
  __shared__ int ll[128];
  int g = blockIdx.x, t = threadIdx.x;
  if (t < 128) { ll[t] = labels[g * 128 + t]; excl[g * 128 + t] = -1; }
  __syncthreads();
  int cnt = 0;
  for (int j = 0; j < 128; ++j) cnt += (ll[j] == t);
  hist[g * 256 + t] = cnt;
  if (t < 128) {
    int mine = ll[t], r = 0;
    for (int j = 0; j < t; ++j) r += (ll[j] == mine);
    lrank[g * 128 + t] = r;
  }
}

// K3: exclusive prefix over chunks per class (in place), write counts
__global__ void k_prefix(int* __restrict__ hist, float* __restrict__ cntf) {
  int c = threadIdx.x;
  int run = 0;
  for (int g = 0; g < 256; ++g) {
    int h = hist[g * 256 + c];
    hist[g * 256 + c] = run;
    run += h;
  }
  cntf[c] = (float)run;
}

// K4: scatter exclusion mapping: member j of class c with rank r serves sample i=r
__global__ void k_scatter(const int* __restrict__ labels, const int* __restrict__ hist,
                          const int* __restrict__ lrank, int* __restrict__ excl) {
  int j = blockIdx.x * blockDim.x + threadIdx.x;
  int c = labels[j];
  int r = hist[(j >> 7) * 256 + c] + lrank[j];
  if (labels[r] == c) excl[r] = j;   // unique (class,rank) -> no write conflicts
}

// K5: partial segment sums via native f32 WMMA:
//   psums[p][c,d] = sum_{b in K-partition p} onehot[c,b] * f_n[b,d]
// grid = 64 d-tiles x 8 K-partitions = 512 blocks; 16 waves = one 16-class tile each
__global__ void __launch_bounds__(512) k_segsum_wmma(const float* __restrict__ feat,
                                                     const int* __restrict__ labels,
                                                     const float* __restrict__ invn,
                                                     float* __restrict__ psums) {
  __shared__ float tb[32 * 16];  // 32 samples x 16 cols, pre-normalized
  __shared__ int   tl[32];
  const int dt = blockIdx.x & 63;
  const int kp = blockIdx.x >> 6;
  const int d0 = dt * 16;
  const int kstart = kp * (B_N / KSPLIT);
  const int kend   = kstart + (B_N / KSPLIT);
  const int t = threadIdx.x;
  const int lane = t & 31;
  const int classBase = (t >> 5) * 16;
  const int m = lane & 15;       // A row (class in tile) == B col (d in tile)
  const int hi = lane >> 4;      // half-wave selects K pair {0,1} vs {2,3}
  const int r = t >> 4, col = t & 15;
  v8f acc = {};
  for (int kb = kstart; kb < kend; kb += 32) {
    int row = kb + r;
    tb[t] = feat[(size_t)row * D_N + d0 + col] * invn[row];
    if (t < 32) tl[t] = labels[kb + t];
    if (kb + 32 < kend)  // prefetch next K-tile (-> global_prefetch_b8)
      __builtin_prefetch(&feat[(size_t)(row + 32) * D_N + d0 + col], 0, 1);
    __syncthreads();
    #pragma unroll
    for (int ks = 0; ks < 8; ++ks) {
      int k0 = ks * 4 + hi * 2;
      int lab0 = tl[k0], lab1 = tl[k0 + 1];
      v2f a, b;
      a.x = (lab0 == classBase + m) ? 1.0f : 0.0f;
      a.y = (lab1 == classBase + m) ? 1.0f : 0.0f;
      b.x = tb[k0 * 16 + m];
      b.y = tb[(k0 + 1) * 16 + m];
      acc = __builtin_amdgcn_wmma_f32_16x16x4_f32(false, a, false, b,
                                                  (short)0, acc, false, false);
    }
    __syncthreads();
  }
  float* out = psums + (size_t)kp * (C_N * D_N);
  #pragma unroll
  for (int v = 0; v < 8; ++v) {
    int mm = hi * 8 + v;  // D layout: VGPR v holds M=v (lanes 0-15) / M=8+v (lanes 16-31)
    out[(size_t)(classBase + mm) * D_N + d0 + m] = acc[v];
  }
}

// K5b: deterministic fixed-order reduction of the 8 K-partials
__global__ void k_reduce_sums(const float* __restrict__ psums, float* __restrict__ sums) {
  int idx = blockIdx.x * blockDim.x + threadIdx.x;  // 0 .. 256*1024-1
  float s = 0.f;
  #pragma unroll
  for (int p = 0; p < KSPLIT; ++p) s += psums[(size_t)p * (C_N * D_N) + idx];
  sums[idx] = s;
}

// K6: per-sample MSE vs leave-one-out centroid; one wave per sample; fixed-order block partial
__global__ void k_loss(const float* __restrict__ feat, const int* __restrict__ labels,
                       const float* __restrict__ invn, const float* __restrict__ sums,
                       const float* __restrict__ cntf, const int* __restrict__ excl,
                       float* __restrict__ partials) {
  __shared__ float wsum[8];
  int wv = threadIdx.x >> 5, lane = threadIdx.x & 31;
  int i = blockIdx.x * 8 + wv;
  int c = labels[i];
  float nf = cntf[c];
  int k = excl[i];
  float inv_i = invn[i];
  const float* fi = feat + (size_t)i * D_N;
  const float* sc = sums + (size_t)c * D_N;
  float acc = 0.f;
  if (k >= 0) {
    float invd = 1.0f / fmaxf(nf - 1.0f, 1.0f);
    float inv_k = invn[k];
    const float* fk = feat + (size_t)k * D_N;
    #pragma unroll
    for (int j = 0; j < 8; ++j) {
      int d = j * 128 + lane * 4;
      float4 a = *(const float4*)(fi + d);
      float4 e = *(const float4*)(fk + d);
      float4 s = *(const float4*)(sc + d);
      float df;
      df = a.x * inv_i - (s.x - e.x * inv_k) * invd; acc += df * df;
      df = a.y * inv_i - (s.y - e.y * inv_k) * invd; acc += df * df;
      df = a.z * inv_i - (s.z - e.z * inv_k) * invd; acc += df * df;
      df = a.w * inv_i - (s.w - e.w * inv_k) * invd; acc += df * df;
    }
  } else {
    float invd = 1.0f / nf;
    #pragma unroll
    for (int j = 0; j < 8; ++j) {
      int d = j * 128 + lane * 4;
      float4 a = *(const float4*)(fi + d);
      float4 s = *(const float4*)(sc + d);
      float df;
      df = a.x * inv_i - s.x * invd; acc += df * df;
      df = a.y * inv_i - s.y * invd; acc += df * df;
      df = a.z * inv_i - s.z * invd; acc += df * df;
      df = a.w * inv_i - s.w * invd; acc += df * df;
    }
  }
  acc = wave_reduce_add(acc);
  if (lane == 0) wsum[wv] = (nf > 1.0f) ? acc * (1.0f / (float)D_N) : 0.0f;
  __syncthreads();
  if (threadIdx.x == 0) {
    float s = 0.f;
    #pragma unroll
    for (int w = 0; w < 8; ++w) s += wsum[w];
    partials[blockIdx.x] = s;
  }
}

// K7: deterministic final reduction of 4096 partials
__global__ void k_final(const float* __restrict__ partials, float* __restrict__ out) {
  __shared__ float sm[256];
  int t = threadIdx.x;
  float s = 0.f;
  #pragma unroll
  for (int j = 0; j < 16; ++j) s += partials[j * 256 + t];
  sm[t] = s;
  __syncthreads();
  for (int st = 128; st >= 1; st >>= 1) {
    if (t < st) sm[t] += sm[t + st];
    __syncthreads();
  }
  if (t == 0) out[0] = W_LOSS * sm[0] / (float)B_N;
}

extern "C" void kernel_launch(void* const* d_in, const int* in_sizes, int n_in,
                              void* d_out, int out_size, void* d_ws, size_t ws_size,
                              hipStream_t stream) {
  const float* feat   = (const float*)d_in[0];
  const int*   labels = (const int*)d_in[1];
  char* ws = (char*)d_ws;
  float* invn  = (float*)(ws + OFF_INVN);
  float* sums  = (float*)(ws + OFF_SUMS);
  int*   hist  = (int*)(ws + OFF_HIST);
  int*   lrank = (int*)(ws + OFF_LRANK);
  int*   excl  = (int*)(ws + OFF_EXCL);
  float* cntf  = (float*)(ws + OFF_CNT);
  float* part  = (float*)(ws + OFF_PART);
  float* psums = (float*)(ws + OFF_PSUM);
  float* out   = (float*)d_out;

  k_norms       <<<B_N / 8, 256, 0, stream>>>(feat, invn);
  k_hist        <<<B_N / 128, 256, 0, stream>>>(labels, hist, lrank, excl);
  k_prefix      <<<1, 256, 0, stream>>>(hist, cntf);
  k_scatter     <<<B_N / 256, 256, 0, stream>>>(labels, hist, lrank, excl);
  k_segsum_wmma <<<(D_N / 16) * KSPLIT, 512, 0, stream>>>(feat, labels, invn, psums);
  k_reduce_sums <<<(C_N * D_N) / 256, 256, 0, stream>>>(psums, sums);
  k_loss        <<<B_N / 8, 256, 0, stream>>>(feat, labels, invn, sums, cntf, excl, part);
  k_final       <<<1, 256, 0, stream>>>(part, out);
}